// DURIS_block_90391881712209
// MI455X (gfx1250) — compile-verified
//
#include <hip/hip_runtime.h>
#include <cstdint>

#define N_PHI 64
#define NKU   16
#define NSU   8
#define BB    512

typedef __attribute__((ext_vector_type(2))) float v2f;
typedef __attribute__((ext_vector_type(8))) float v8f;
typedef __attribute__((ext_vector_type(4))) unsigned int u32x4;
typedef __attribute__((ext_vector_type(4))) int i32x4;
typedef __attribute__((ext_vector_type(8))) int i32x8;

#define BT_STR 68          // transposed B operand row stride (68 % 64 == 4 -> b64 frags conflict-free)
#define PSTR   68          // P result tile stride (8*68 % 64 == 32 -> conflict-free acc stores)
// Pair-interleaved A pack: the two K-values a lane needs are adjacent -> one ds_load_b64
#define APACK(k, r) ((((k) >> 1) << 5) + (((r)) << 1) + ((k) & 1))

#if __has_builtin(__builtin_amdgcn_tensor_load_to_lds) && __has_builtin(__builtin_amdgcn_s_wait_tensorcnt)
#define DURIS_TDM 1
#else
#define DURIS_TDM 0
#endif

// ---------------- Kernel 1: learned step size (row sums -> BN -> dense -> leaky relu) ----
__global__ __launch_bounds__(128) void duris_step(
    const float* __restrict__ PhiRe, const float* __restrict__ PhiIm,
    const float* __restrict__ gam, const float* __restrict__ bet,
    const float* __restrict__ mean, const float* __restrict__ var,
    const float* __restrict__ w, const float* __restrict__ bias,
    float* __restrict__ step_out)
{
  const int b = blockIdx.x;
  const int t = threadIdx.x;
  const float* src = (t < 64) ? (PhiRe + (size_t)b*4096 + (size_t)t*64)
                              : (PhiIm + (size_t)b*4096 + (size_t)(t-64)*64);
  float acc = 0.f;
  #pragma unroll 8
  for (int j = 0; j < 64; ++j) acc += src[j];
  const float tn = gam[t]*(acc - mean[t])*rsqrtf(var[t] + 1e-3f) + bet[t];
  __shared__ float red[128];
  red[t] = tn * w[t];
  __syncthreads();
  for (int off = 64; off > 0; off >>= 1) {
    if (t < off) red[t] += red[t + off];
    __syncthreads();
  }
  if (t == 0) {
    const float s = red[0] + bias[0];
    step_out[b] = (s >= 0.f) ? s : 0.1f * s;
  }
}

// ---------------- parallel 8x8 complex Gauss-Jordan on [A|I] (aug 8x16, 128 threads) -----
__device__ __forceinline__ void gauss_jordan8(float* augRe, float* augIm, int t)
{
  const int r = t >> 4, c = t & 15;
  for (int i = 0; i < 8; ++i) {
    __syncthreads();                       // prior iteration's writes visible
    const float prr = augRe[i*16 + i], pii = augIm[i*16 + i];
    const float dinv = 1.f / (prr*prr + pii*pii);
    const float qr = prr*dinv, qi = -pii*dinv;
    __syncthreads();                       // pivot read by all before row-i rescale
    if (t < 128 && r == i) {
      const float ar = augRe[t], ai = augIm[t];
      augRe[t] = ar*qr - ai*qi;
      augIm[t] = ar*qi + ai*qr;
    }
    __syncthreads();
    float fr = 0.f, fi = 0.f, gr = 0.f, gi = 0.f;
    if (t < 128) {
      fr = augRe[r*16 + i]; fi = augIm[r*16 + i];
      gr = augRe[i*16 + c]; gi = augIm[i*16 + c];
    }
    __syncthreads();                       // capture factors before elimination writes
    if (t < 128 && r != i) {
      augRe[t] -= fr*gr - fi*gi;
      augIm[t] -= fr*gi + fi*gr;
    }
  }
  __syncthreads();
}

// ---------------- Kernel 2: per-(k,b) heavy pipeline --------------------------------------
__global__ __launch_bounds__(256) void duris_heavy(
    const float* __restrict__ PhiRe, const float* __restrict__ PhiIm,
    const float* __restrict__ GhRe,  const float* __restrict__ GhIm,
    const float* __restrict__ LamRe, const float* __restrict__ LamIm,
    const float* __restrict__ CRe,   const float* __restrict__ CIm,
    const float* __restrict__ Beta,
    float* __restrict__ dK)   // [NK][B][64][2] diagonal contributions
{
  const int b = blockIdx.x;
  const int k = blockIdx.y;
  const int t = threadIdx.x;
  const int lane = t & 31;
  const int wv = t >> 5;

  __shared__ __align__(16) float sPhiRe[64*BT_STR], sPhiIm[64*BT_STR]; // Phi^T (staged)
  __shared__ __align__(16) float sCRe[64*BT_STR],   sCIm[64*BT_STR];   // C row-major (TDM), read transposed
  __shared__ __align__(16) float sPr[16*PSTR],      sPi[16*PSTR];      // 16x64 WMMA results
  __shared__ __align__(16) float sA1[32*32];                           // [G_re;G_im] pack, pair-interleaved
  __shared__ __align__(16) float sA2[32*32];                           // [M_re;M_im] pack
  __shared__ __align__(16) float sT [32*32];                           // [T_re;T_im] pack
  __shared__ float sLamRe[64], sLamIm[64];
  __shared__ float augRe[128], augIm[128];
  __shared__ float qiRe[64], qiIm[64];
  __shared__ float q2Re[64], q2Im[64];
  __shared__ float sInvBeta;

  const size_t kb = (size_t)k*BB + b;
  const float* Cre_g = CRe + kb*4096;
  const float* Cim_g = CIm + kb*4096;

#if DURIS_TDM
  // ---- TDM: async-DMA C[k,b] (2x 16KB) into LDS with hardware row padding (64 dw data,
  //      4 dw pad -> BT_STR=68 layout), overlapping the whole matmul-1 phase. Wave 0 only.
  if (wv == 0) {
    // group1: wg_mask=0 | data_size=4B(2)<<16 | pad_en<<20 | pad_interval(64dw: code5)<<22
    //         | pad_amount(4dw: code3)<<25 ; tensor_dim0=4096 (bits 79:48);
    //         tensor_dim1=1 (bits 111:80); tile_dim0=4096 (bits 127:112);
    //         tile_dim1/2=0 ; tensor_dim0_stride=4096 (bits 207:160)
    const i32x8 g1 = { (int)0x07520000, (int)0x10000000, (int)0x00010000, (int)0x10000000,
                       0, 4096, 0, 0 };
    const i32x4 z4 = { 0, 0, 0, 0 };
    const i32x8 z8 = { 0, 0, 0, 0, 0, 0, 0, 0 };
    const unsigned lds_re = (unsigned)(uintptr_t)(const void*)sCRe;  // flat[31:0] == LDS offset
    const unsigned lds_im = (unsigned)(uintptr_t)(const void*)sCIm;
    const unsigned long long ga_re = (unsigned long long)(uintptr_t)(const void*)Cre_g;
    const unsigned long long ga_im = (unsigned long long)(uintptr_t)(const void*)Cim_g;
    u32x4 g0 = { 1u,                                   // count=1 valid user descriptor
                 lds_re,
                 (unsigned)(ga_re & 0xffffffffu),
                 ((unsigned)((ga_re >> 32) & 0x01ffffffu)) | 0x80000000u };  // type=2
    __builtin_amdgcn_tensor_load_to_lds(g0, g1, z4, z4, z8, 0);
    g0.y = lds_im;
    g0.z = (unsigned)(ga_im & 0xffffffffu);
    g0.w = ((unsigned)((ga_im >> 32) & 0x01ffffffu)) | 0x80000000u;
    __builtin_amdgcn_tensor_load_to_lds(g0, g1, z4, z4, z8, 0);
  }
#else
  // Fallback: L2 prefetch of C while matmul 1 runs (global_prefetch_b8)
  {
    const float* p = (t < 128) ? (Cre_g + (size_t)t*32) : (Cim_g + (size_t)(t-128)*32);
    __builtin_prefetch(p, 0, 1);
  }
#endif

  // ---- stage Phi TRANSPOSED (coalesced global reads), G pack, Lam, Beta ----
  const size_t b4096 = (size_t)b*4096;
  for (int idx = t; idx < 4096; idx += 256) {
    const int e = idx >> 6, c = idx & 63;        // Phi[e][c] -> sPhiT[c][e]
    sPhiRe[c*BT_STR + e] = PhiRe[b4096 + idx];
    sPhiIm[c*BT_STR + e] = PhiIm[b4096 + idx];
  }
  const float* Gre_g = GhRe + kb*512;   // G_h[k,b]: [64][8]
  const float* Gim_g = GhIm + kb*512;
  for (int idx = t; idx < 512; idx += 256) {
    const int e = idx >> 3, s = idx & 7;
    sA1[APACK(e, s)]     =  Gre_g[idx];   // G_re[s][e] = G_h_re[e][s]
    sA1[APACK(e, s + 8)] = -Gim_g[idx];   // G_im[s][e] = -G_h_im[e][s]
  }
  if (t < 64) {
    sLamRe[t] = LamRe[kb*64 + t];
    sLamIm[t] = LamIm[kb*64 + t];
  }
  if (t == 0) sInvBeta = 1.f / Beta[kb];
  __syncthreads();

  // Wave job: bsel selects {re,im} B matrix, ntile selects 16-wide N tile
  const int r16   = lane & 15;
  const int kh    = (lane >> 4) * 2;      // A/B fragment K offset for lanes 16..31
  const int bsel  = wv >> 2;
  const int ntile = wv & 3;
  const int cN    = ntile*16 + r16;
  const int mrow  = (lane >> 4) * 8;

  // ---- matmul 1: P = [G_re;G_im](16x64) @ Phi_{re,im}(64x64) via v_wmma_f32_16x16x4_f32
  {
    const float* sB = bsel ? sPhiIm : sPhiRe;
    v8f acc = {0.f,0.f,0.f,0.f,0.f,0.f,0.f,0.f};
    #pragma unroll
    for (int ki = 0; ki < 16; ++ki) {
      const int k0 = ki*4 + kh;                       // even
      const v2f a  = *reinterpret_cast<const v2f*>(&sA1[((k0 >> 1) << 5) + (r16 << 1)]);
      const v2f bb = *reinterpret_cast<const v2f*>(&sB [cN*BT_STR + k0]);
      acc = __builtin_amdgcn_wmma_f32_16x16x4_f32(false, a, false, bb,
                                                  (short)0, acc, false, false);
    }
    float* dst = bsel ? sPi : sPr;
    #pragma unroll
    for (int v = 0; v < 8; ++v) dst[(mrow + v)*PSTR + cN] = acc[v];
  }
  __syncthreads();

  // ---- combine -> M pack (A2, pair-interleaved) ----
  for (int idx = t; idx < 512; idx += 256) {
    const int s = idx >> 6, f = idx & 63;
    const float mre = sPr[s*PSTR + f] - sPi[(s+8)*PSTR + f];
    const float mim = sPi[s*PSTR + f] + sPr[(s+8)*PSTR + f];
    sA2[APACK(f, s)]     = mre;
    sA2[APACK(f, s + 8)] = mim;
  }
#if DURIS_TDM
  if (wv == 0) __builtin_amdgcn_s_wait_tensorcnt((short)0);  // C tiles landed in LDS
#else
  for (int idx = t; idx < 4096; idx += 256) {
    const int r = idx >> 6, c = idx & 63;
    sCRe[r*BT_STR + c] = Cre_g[idx];
    sCIm[r*BT_STR + c] = Cim_g[idx];
  }
#endif
  __syncthreads();

  // ---- matmul 2: C is Hermitian (exact in fp32), read row-major C transposed:
  //      C_re^T == C_re, C_im^T == -C_im -> signs folded into the T-combine below.
  {
    const float* sB = bsel ? sCIm : sCRe;
    v8f acc = {0.f,0.f,0.f,0.f,0.f,0.f,0.f,0.f};
    #pragma unroll
    for (int ki = 0; ki < 16; ++ki) {
      const int k0 = ki*4 + kh;                       // even
      const v2f a  = *reinterpret_cast<const v2f*>(&sA2[((k0 >> 1) << 5) + (r16 << 1)]);
      const v2f bb = *reinterpret_cast<const v2f*>(&sB [cN*BT_STR + k0]);
      acc = __builtin_amdgcn_wmma_f32_16x16x4_f32(false, a, false, bb,
                                                  (short)0, acc, false, false);
    }
    float* dst = bsel ? sPi : sPr;                    // sPr = U_r ; sPi = W = -U_i
    #pragma unroll
    for (int v = 0; v < 8; ++v) dst[(mrow + v)*PSTR + cN] = acc[v];
  }
  __syncthreads();

  // ---- combine -> T pack (sign-corrected: T_re = U_r + W[8:], T_im = U_r[8:] - W) ----
  for (int idx = t; idx < 512; idx += 256) {
    const int s = idx >> 6, f = idx & 63;
    const float tre = sPr[s*PSTR + f] + sPi[(s+8)*PSTR + f];
    const float tim = sPr[(s+8)*PSTR + f] - sPi[s*PSTR + f];
    sT[APACK(f, s)]     = tre;
    sT[APACK(f, s + 8)] = tim;
  }

  // ---- Linv = inv(Lam) ----
  if (t < 128) {
    const int r = t >> 4, c = t & 15;
    augRe[t] = (c < 8) ? sLamRe[r*8 + c] : ((c - 8 == r) ? 1.f : 0.f);
    augIm[t] = (c < 8) ? sLamIm[r*8 + c] : 0.f;
  }
  __syncthreads();
  gauss_jordan8(augRe, augIm, t);
  if (t < 64) {                           // qi* temporarily holds Linv
    qiRe[t] = augRe[(t >> 3)*16 + 8 + (t & 7)];
    qiIm[t] = augIm[(t >> 3)*16 + 8 + (t & 7)];
  }
  __syncthreads();

  // ---- Q = Linv + (T @ M^H) * invBeta ; Qi = inv(Q) ----
  if (t < 64) {
    const int r = t >> 3, c = t & 7;
    float sr = 0.f, si = 0.f;
    for (int f = 0; f < 64; ++f) {
      const float tr = sT [APACK(f, r)], ti = sT [APACK(f, r + 8)];
      const float mr = sA2[APACK(f, c)], mi = sA2[APACK(f, c + 8)];
      sr += tr*mr + ti*mi;                // T * conj(M)
      si += ti*mr - tr*mi;
    }
    const float ib = sInvBeta;
    augRe[r*16 + c] = qiRe[t] + sr*ib;
    augIm[r*16 + c] = qiIm[t] + si*ib;
    augRe[r*16 + 8 + c] = (r == c) ? 1.f : 0.f;
    augIm[r*16 + 8 + c] = 0.f;
  }
  __syncthreads();
  gauss_jordan8(augRe, augIm, t);
  if (t < 64) {
    qiRe[t] = augRe[(t >> 3)*16 + 8 + (t & 7)];
    qiIm[t] = augIm[(t >> 3)*16 + 8 + (t & 7)];
  }
  __syncthreads();

  // ---- Qi2 = Qi @ Qi ----
  if (t < 64) {
    const int r = t >> 3, c = t & 7;
    float ar = 0.f, ai = 0.f;
    #pragma unroll
    for (int m = 0; m < 8; ++m) {
      const float x = qiRe[r*8 + m], y = qiIm[r*8 + m];
      const float u = qiRe[m*8 + c], v = qiIm[m*8 + c];
      ar += x*u - y*v;
      ai += x*v + y*u;
    }
    q2Re[t] = ar; q2Im[t] = ai;
  }
  __syncthreads();

  // ---- d[f] = (G_h @ Qi2 @ T)[f][f] * invBeta  (only the diagonal survives eye-mask) ----
  if (t < 64) {
    const int f = t;
    float dr = 0.f, di = 0.f;
    #pragma unroll
    for (int tt = 0; tt < 8; ++tt) {
      float xr = 0.f, xi = 0.f;
      #pragma unroll
      for (int s = 0; s < 8; ++s) {
        const float gr =  sA1[APACK(f, s)];        // G_h_re[f][s]
        const float gi = -sA1[APACK(f, s + 8)];    // G_h_im[f][s]
        const float qr = q2Re[s*8 + tt], qi = q2Im[s*8 + tt];
        xr += gr*qr - gi*qi;
        xi += gr*qi + gi*qr;
      }
      const float tr = sT[APACK(f, tt)], ti = sT[APACK(f, tt + 8)];  // T[tt][f]
      dr += xr*tr - xi*ti;
      di += xr*ti + xi*tr;
    }
    const float ib = sInvBeta;
    dK[(kb*64 + f)*2 + 0] = dr*ib;
    dK[(kb*64 + f)*2 + 1] = di*ib;
  }
}

// ---------------- Kernel 3: finalize (diag update + elementwise clamp, interleaved out) --
__global__ __launch_bounds__(256) void duris_finalize(
    const float* __restrict__ PhiRe, const float* __restrict__ PhiIm,
    const float* __restrict__ dK, const float* __restrict__ stepArr,
    float* __restrict__ out)
{
  const int b = blockIdx.x;
  const int t = threadIdx.x;
  __shared__ float rgRe[64], rgIm[64], red[64];
  __shared__ float sScale;
  const size_t b4096 = (size_t)b*4096;
  if (t < 64) {
    float er = 0.f, ei = 0.f;
    for (int k = 0; k < NKU; ++k) {         // fixed-order reduction (deterministic)
      const size_t idx = (((size_t)k*BB + b)*64 + t)*2;
      er += dK[idx]; ei += dK[idx + 1];
    }
    er *= -(1.f/NKU); ei *= -(1.f/NKU);
    const float pr = PhiRe[b4096 + (size_t)t*65];
    const float pi = PhiIm[b4096 + (size_t)t*65];
    const float proj = er*pr + ei*pi;       // Re(eg * conj(Phi)) on the diagonal
    const float rr = er - proj*pr;
    const float ri = ei - proj*pi;
    rgRe[t] = rr; rgIm[t] = ri;
    red[t] = rr*rr + ri*ri;
  }
  __syncthreads();
  if (t == 0) {
    float s = 0.f;
    for (int i = 0; i < 64; ++i) s += red[i];
    sScale = stepArr[b] / sqrtf(s);
  }
  __syncthreads();
  const float scale = sScale;
  for (int idx = t; idx < 4096; idx += 256) {
    const int i = idx >> 6, j = idx & 63;
    float pr = PhiRe[b4096 + idx];
    float pi = PhiIm[b4096 + idx];
    if (i == j) { pr -= scale*rgRe[i]; pi -= scale*rgIm[i]; }
    const float ab = sqrtf(pr*pr + pi*pi);
    const float den = fmaxf(ab - 1.f, 0.f) + 1.f;
    float2 o; o.x = pr/den; o.y = pi/den;
    reinterpret_cast<float2*>(out)[b4096 + idx] = o;   // complex64 interleaved
  }
}

extern "C" void kernel_launch(void* const* d_in, const int* in_sizes, int n_in,
                              void* d_out, int out_size, void* d_ws, size_t ws_size,
                              hipStream_t stream)
{
  (void)in_sizes; (void)n_in; (void)out_size; (void)ws_size;
  const float* PhiRe = (const float*)d_in[0];
  const float* PhiIm = (const float*)d_in[1];
  const float* GhRe  = (const float*)d_in[2];
  const float* GhIm  = (const float*)d_in[3];
  const float* LamRe = (const float*)d_in[4];
  const float* LamIm = (const float*)d_in[5];
  const float* CRe   = (const float*)d_in[6];
  const float* CIm   = (const float*)d_in[7];
  const float* Beta  = (const float*)d_in[8];
  const float* gam   = (const float*)d_in[9];
  const float* bet   = (const float*)d_in[10];
  const float* mean  = (const float*)d_in[11];
  const float* var   = (const float*)d_in[12];
  const float* w     = (const float*)d_in[13];
  const float* bias  = (const float*)d_in[14];

  float* ws   = (float*)d_ws;
  float* step = ws;          // [512]
  float* dK   = ws + 512;    // [16][512][64][2] = 4 MB; fully overwritten each call

  duris_step<<<dim3(BB), dim3(128), 0, stream>>>(
      PhiRe, PhiIm, gam, bet, mean, var, w, bias, step);
  duris_heavy<<<dim3(BB, NKU), dim3(256), 0, stream>>>(
      PhiRe, PhiIm, GhRe, GhIm, LamRe, LamIm, CRe, CIm, Beta, dK);
  duris_finalize<<<dim3(BB), dim3(256), 0, stream>>>(
      PhiRe, PhiIm, dK, step, (float*)d_out);
}